// Embedding_20040317403544
// MI455X (gfx1250) — compile-verified
//
#include <hip/hip_runtime.h>

// Embedding row-gather: out[r, :] = w[tok[r], :], r in [0, B*S), D_MODEL=512 fp32.
// Pure data movement (0 FLOPs): ~64 MiB total traffic -> bandwidth-roofline op.
// CDNA5 path: Async LDS engine (ASYNCcnt) moves each 2KB row memory->LDS->memory
// without touching VGPRs; wave only issues addresses. Double-buffered LDS staging
// per wave; global_prefetch_b8 pulls the next gathered row toward L2.

#define D_MODEL         512
#define ROW_BYTES       (D_MODEL * 4)      // 2048 B per row
#define WAVES_PER_BLOCK 8                  // 256 threads, wave32
#define BUFS_PER_WAVE   2                  // double buffer

__global__ __launch_bounds__(256) void embed_gather_async_kernel(
    const int* __restrict__ tok,
    const float* __restrict__ w,
    float* __restrict__ out,
    int n_rows,
    int rows_per_wave)
{
    // 8 waves * 2 bufs * 2KB = 32KB LDS per block (WGP has 320KB).
    __shared__ __align__(16) char smem[WAVES_PER_BLOCK * BUFS_PER_WAVE * ROW_BYTES];

    const int tid     = blockIdx.x * blockDim.x + threadIdx.x;
    const int wave    = tid >> 5;                 // global wave id (wave32)
    const int lane    = tid & 31;
    const int wib     = threadIdx.x >> 5;         // wave index within block
    const int n_waves = (gridDim.x * blockDim.x) >> 5;

    // Per-lane LDS byte address of this wave's buffer 0 (low 32 bits of the
    // flat address of an LDS object are its LDS byte offset).
    const unsigned lds0 =
        (unsigned)(size_t)(&smem[wib * (BUFS_PER_WAVE * ROW_BYTES)]) +
        (unsigned)(lane * 16);

    for (int k = 0; k < rows_per_wave; ++k) {
        const int r = wave + k * n_waves;         // wave-uniform -> uniform branch
        if (r >= n_rows) break;

        // Token id is wave-uniform: force into an SGPR so the row base is scalar.
        const int t = __builtin_amdgcn_readfirstlane(tok[r]);

        const unsigned lds = lds0 + (unsigned)((k & 1) * ROW_BYTES);
        const unsigned long long gsrc =
            (unsigned long long)(const char*)(w + (size_t)t * D_MODEL) + (unsigned)(lane * 16);
        const unsigned long long gdst =
            (unsigned long long)(char*)(out + (size_t)r * D_MODEL) + (unsigned)(lane * 16);

        // Prefetch next gathered row toward L2 while this one streams
        // (one byte per 64B per lane covers the full 2KB row).
        const int rn = r + n_waves;
        if ((k + 1 < rows_per_wave) && (rn < n_rows)) {
            const int tn = __builtin_amdgcn_readfirstlane(tok[rn]);
            __builtin_prefetch((const char*)(w + (size_t)tn * D_MODEL) + lane * 64, 0, 3);
        }

        // Async gather: memory -> LDS. INST_OFFSET applies to both the LDS and
        // global addresses, so one base covers the whole 2KB row.
        asm volatile("global_load_async_to_lds_b128 %0, %1, off"             :: "v"(lds), "v"(gsrc) : "memory");
        asm volatile("global_load_async_to_lds_b128 %0, %1, off offset:512"  :: "v"(lds), "v"(gsrc) : "memory");
        asm volatile("global_load_async_to_lds_b128 %0, %1, off offset:1024" :: "v"(lds), "v"(gsrc) : "memory");
        asm volatile("global_load_async_to_lds_b128 %0, %1, off offset:1536" :: "v"(lds), "v"(gsrc) : "memory");

        // Drain this row's loads (also drains the previous row's stores, so the
        // other buffer is free for the next iteration -> no LDS WAR hazard).
        asm volatile("s_wait_asynccnt 0x0" ::: "memory");

        // Async write-out: LDS -> memory (overlaps with next iteration's loads).
        asm volatile("global_store_async_from_lds_b128 %0, %1, off"             :: "v"(gdst), "v"(lds) : "memory");
        asm volatile("global_store_async_from_lds_b128 %0, %1, off offset:512"  :: "v"(gdst), "v"(lds) : "memory");
        asm volatile("global_store_async_from_lds_b128 %0, %1, off offset:1024" :: "v"(gdst), "v"(lds) : "memory");
        asm volatile("global_store_async_from_lds_b128 %0, %1, off offset:1536" :: "v"(gdst), "v"(lds) : "memory");
    }
    // s_endpgm performs an implicit wait-idle, draining the final async stores.
}

extern "C" void kernel_launch(void* const* d_in, const int* in_sizes, int n_in,
                              void* d_out, int out_size, void* d_ws, size_t ws_size,
                              hipStream_t stream) {
    const int*   tok = (const int*)d_in[0];    // (B, S) int32 token ids
    const float* w   = (const float*)d_in[1];  // (VOCAB, D_MODEL) fp32 table
    float*       out = (float*)d_out;          // (B, S, D_MODEL) fp32

    const int n_rows = in_sizes[0];            // B*S = 16384

    const int block   = 256;                   // 8 waves per block
    const int blocks  = 512;                   // 4096 waves total
    const int n_waves = blocks * (block / 32);
    const int rows_per_wave = (n_rows + n_waves - 1) / n_waves;  // = 4

    embed_gather_async_kernel<<<blocks, block, 0, stream>>>(tok, w, out, n_rows, rows_per_wave);
}